// GATFormulaEncoder_88794153877800
// MI455X (gfx1250) — compile-verified
//
#include <hip/hip_runtime.h>
#include <hip/hip_bf16.h>
#include <math.h>

#define NN   20000
#define EE   320000
#define ETOT (EE + NN)
#define GG   128
#define DE   64
#define HH   4
#define CC   128
#define DD   512
#define OUTD 256

typedef __attribute__((ext_vector_type(2))) float v2f;
typedef __attribute__((ext_vector_type(8))) float v8f;

// ---------------- utility kernels ----------------
__global__ void k_fill_f32(float* p, float v, int n) {
  int i = blockIdx.x * blockDim.x + threadIdx.x;
  if (i < n) p[i] = v;
}
__global__ void k_fill_i32(int* p, int v, int n) {
  int i = blockIdx.x * blockDim.x + threadIdx.x;
  if (i < n) p[i] = v;
}
__global__ void k_copy_i32(int* dst, const int* src, int n) {
  int i = blockIdx.x * blockDim.x + threadIdx.x;
  if (i < n) dst[i] = src[i];
}

// ---------------- embedding gather ----------------
__global__ void k_gather_emb(const int* __restrict__ ids, const float* __restrict__ emb,
                             float* __restrict__ x0) {
  int i = blockIdx.x * blockDim.x + threadIdx.x;
  if (i >= NN * DE) return;
  int n = i >> 6, c = i & 63;
  x0[i] = emb[ids[n] * DE + c];
}

// ---------------- CSR build (dst-sorted) ----------------
__global__ void k_count_deg(const int* __restrict__ ei, int* __restrict__ counts) {
  int e = blockIdx.x * blockDim.x + threadIdx.x;
  if (e >= ETOT) return;
  int d = (e < EE) ? ei[EE + e] : (e - EE);
  atomicAdd(&counts[d], 1);
}
__global__ void k_scan(const int* __restrict__ counts, int* __restrict__ indptr) {
  __shared__ int part[1024];
  int t = threadIdx.x;
  const int CH = (NN + 1023) / 1024;
  int base = t * CH;
  int s = 0;
  for (int i = 0; i < CH && base + i < NN; ++i) s += counts[base + i];
  part[t] = s;
  __syncthreads();
  if (t == 0) {
    int run = 0;
    for (int i = 0; i < 1024; ++i) { int v = part[i]; part[i] = run; run += v; }
    indptr[NN] = run;
  }
  __syncthreads();
  int run = part[t];
  for (int i = 0; i < CH && base + i < NN; ++i) {
    indptr[base + i] = run;
    run += counts[base + i];
  }
}
__global__ void k_fill_csr(const int* __restrict__ ei, int* __restrict__ cursor,
                           int* __restrict__ csr_src) {
  int e = blockIdx.x * blockDim.x + threadIdx.x;
  if (e >= ETOT) return;
  int s = (e < EE) ? ei[e] : (e - EE);
  int d = (e < EE) ? ei[EE + e] : (e - EE);
  int pos = atomicAdd(&cursor[d], 1);
  csr_src[pos] = s;
}
__global__ void k_gcount(const int* __restrict__ batch, int* __restrict__ gcount) {
  int n = blockIdx.x * blockDim.x + threadIdx.x;
  if (n < NN) atomicAdd(&gcount[batch[n]], 1);
}

// ---------------- fp32 WMMA GEMM with async B staging into LDS --------------
// C[M,Nc] = A[M,K] @ B[K,Nc]; M%16==0, K%16==0, Nc%NB==0, NB in {256,512}.
// Block: one 16-row block x NB columns; NB/64 waves (NTHR = NB/2 threads).
// B is staged 16 K-rows at a time into LDS via global_load_async_to_lds_b128.
template <int NB>
__global__ void k_gemm(const float* __restrict__ A, const float* __restrict__ B,
                       float* __restrict__ Cm, int M, int K, int Nc) {
  constexpr int NTHR = NB / 2;                 // 128 or 256 threads
  constexpr int PERTHR = (16 * NB / 4) / NTHR; // 8 float4 slots per thread
  __shared__ float bsh[16 * NB];
  int m0  = blockIdx.x * 16;
  int cb0 = blockIdx.y * NB;
  int t = threadIdx.x;
  int lane = t & 31;
  int n0 = (t >> 5) * 64;                      // wave's 64-col strip within block
  int arow  = lane & 15;
  int khalf = (lane >> 4) << 1;                // ISA 16x4 A layout: 0 or 2
  int bcol  = lane & 15;
  v8f c0 = {}, c1 = {}, c2 = {}, c3 = {};
  for (int k0 = 0; k0 < K; k0 += 16) {
    // --- stage B[k0..k0+15, cb0..cb0+NB) into LDS (async, coalesced b128) ---
    #pragma unroll
    for (int j = 0; j < PERTHR; ++j) {
      int vid = t + j * NTHR;
      int r  = (vid * 4) / NB;                 // constexpr NB -> shift
      int cc = (vid * 4) & (NB - 1);
      const float* gp = B + (size_t)(k0 + r) * Nc + cb0 + cc;
      // low 32 bits of a generic LDS pointer == wave-relative LDS byte address;
      // ptrtoint also marks bsh as escaped so the asm's memory clobber is honored.
      unsigned ldsOff = (unsigned)(size_t)&bsh[r * NB + cc];
      asm volatile("global_load_async_to_lds_b128 %0, %1, off"
                   :: "v"(ldsOff), "v"((unsigned long long)(size_t)gp)
                   : "memory");
    }
    asm volatile("s_wait_asynccnt 0x0" ::: "memory");
    __syncthreads();
    // --- 4 k-steps of 4 WMMAs, B fragments from LDS ---
    #pragma unroll
    for (int kk = 0; kk < 16; kk += 4) {
      const float* ap = A + (size_t)(m0 + arow) * K + k0 + kk + khalf;
      v2f a; a.x = ap[0]; a.y = ap[1];
      const float* bp  = &bsh[(kk + khalf) * NB + n0 + bcol];
      const float* bp1 = bp + NB;
      v2f b0; b0.x = bp[0];  b0.y = bp1[0];
      v2f b1; b1.x = bp[16]; b1.y = bp1[16];
      v2f b2; b2.x = bp[32]; b2.y = bp1[32];
      v2f b3; b3.x = bp[48]; b3.y = bp1[48];
      c0 = __builtin_amdgcn_wmma_f32_16x16x4_f32(false, a, false, b0, (short)0, c0, false, false);
      c1 = __builtin_amdgcn_wmma_f32_16x16x4_f32(false, a, false, b1, (short)0, c1, false, false);
      c2 = __builtin_amdgcn_wmma_f32_16x16x4_f32(false, a, false, b2, (short)0, c2, false, false);
      c3 = __builtin_amdgcn_wmma_f32_16x16x4_f32(false, a, false, b3, (short)0, c3, false, false);
    }
    __syncthreads();
  }
  int crow = (lane >> 4) << 3;
  int ccol = lane & 15;
  float* cp = Cm + (size_t)(m0 + crow) * Nc + cb0 + n0 + ccol;
  #pragma unroll
  for (int r = 0; r < 8; ++r) {
    cp[(size_t)r * Nc +  0] = c0[r];
    cp[(size_t)r * Nc + 16] = c1[r];
    cp[(size_t)r * Nc + 32] = c2[r];
    cp[(size_t)r * Nc + 48] = c3[r];
  }
}

// ---------------- attention scores: one wave per (node, head) ----------------
__global__ void k_scores(const float* __restrict__ xp, const float* __restrict__ av,
                         const float* __restrict__ adv, float* __restrict__ ssrc,
                         float* __restrict__ sdst) {
  int w = blockIdx.x * (blockDim.x >> 5) + (threadIdx.x >> 5);
  if (w >= NN * HH) return;
  int n = w >> 2, h = w & 3, lane = threadIdx.x & 31;
  const float4 xv = *(const float4*)(xp + (size_t)n * DD + h * CC + lane * 4);
  const float4 a1 = *(const float4*)(av  + h * CC + lane * 4);
  const float4 a2 = *(const float4*)(adv + h * CC + lane * 4);
  float s = xv.x * a1.x + xv.y * a1.y + xv.z * a1.z + xv.w * a1.w;
  float d = xv.x * a2.x + xv.y * a2.y + xv.z * a2.z + xv.w * a2.w;
  for (int m = 16; m; m >>= 1) { s += __shfl_xor(s, m); d += __shfl_xor(d, m); }
  if (lane == 0) { ssrc[w] = s; sdst[w] = d; }
}

// ---------------- GAT aggregation: block per dst node, wave per head ----------
__global__ void k_aggregate(const float* __restrict__ xp, const float* __restrict__ ssrc,
                            const float* __restrict__ sdst, const int* __restrict__ indptr,
                            const int* __restrict__ csr_src, float* __restrict__ agg) {
  int n = blockIdx.x;
  int h = threadIdx.x >> 5;     // 4 waves = 4 heads
  int lane = threadIdx.x & 31;
  int beg = indptr[n], end = indptr[n + 1];
  float sd = sdst[n * HH + h];
  // pass 1: running max of leaky-relu(e)
  float m = -INFINITY;
  for (int i = beg; i < end; ++i) {
    float v = ssrc[csr_src[i] * HH + h] + sd;
    v = (v > 0.0f) ? v : v * 0.2f;
    m = fmaxf(m, v);
  }
  // pass 2: exp-weighted accumulate (softmax fused)
  float a0 = 0.f, a1 = 0.f, a2 = 0.f, a3 = 0.f, denom = 0.f;
  for (int i = beg; i < end; ++i) {
    int s = csr_src[i];
    float v = ssrc[s * HH + h] + sd;
    v = (v > 0.0f) ? v : v * 0.2f;
    float ex = __expf(v - m);
    denom += ex;
    const float4 xv = *(const float4*)(xp + (size_t)s * DD + h * CC + lane * 4);
    a0 += ex * xv.x; a1 += ex * xv.y; a2 += ex * xv.z; a3 += ex * xv.w;
  }
  float inv = 1.0f / denom;   // self-loop guarantees denom >= 1
  float4 o = { a0 * inv, a1 * inv, a2 * inv, a3 * inv };
  *(float4*)(agg + (size_t)n * DD + h * CC + lane * 4) = o;
}

// ---------------- x = agg + bias + residual; per-graph sum/sumsq -------------
__global__ void k_combine_stats(const float* __restrict__ agg, const float* __restrict__ bias,
                                const float* __restrict__ res, const int* __restrict__ batch,
                                float* __restrict__ xd, float* __restrict__ gsum,
                                float* __restrict__ gsq) {
  __shared__ float sh[16];
  int n = blockIdx.x, t = threadIdx.x;
  int lane = t & 31, w = t >> 5;
  size_t base = (size_t)n * DD;
  float v0 = agg[base + t]       + bias[t]       + res[base + t];
  float v1 = agg[base + t + 256] + bias[t + 256] + res[base + t + 256];
  xd[base + t] = v0;
  xd[base + t + 256] = v1;
  float s = v0 + v1, q = v0 * v0 + v1 * v1;
  for (int m = 16; m; m >>= 1) { s += __shfl_xor(s, m); q += __shfl_xor(q, m); }
  if (lane == 0) { sh[w] = s; sh[8 + w] = q; }
  __syncthreads();
  if (t == 0) {
    float S = 0.f, Q = 0.f;
    for (int i = 0; i < 8; ++i) { S += sh[i]; Q += sh[8 + i]; }
    int g = batch[n];
    atomicAdd(&gsum[g], S);
    atomicAdd(&gsq[g], Q);
  }
}
__global__ void k_ln_params(const float* __restrict__ gsum, const float* __restrict__ gsq,
                            const int* __restrict__ gcount, float* __restrict__ gmu,
                            float* __restrict__ grs) {
  int g = threadIdx.x;
  if (g >= GG) return;
  float cnt = fmaxf((float)gcount[g] * (float)DD, 1.0f);
  float mu = gsum[g] / cnt;
  float var = fmaxf(gsq[g] / cnt - mu * mu, 0.0f);
  gmu[g] = mu;
  grs[g] = rsqrtf(var + 1e-5f);
}
__global__ void k_ln_apply(float* __restrict__ x, const int* __restrict__ batch,
                           const float* __restrict__ gmu, const float* __restrict__ grs,
                           const float* __restrict__ lw, const float* __restrict__ lb) {
  int n = blockIdx.x, t = threadIdx.x;
  int g = batch[n];
  float mu = gmu[g], rs = grs[g];
  size_t base = (size_t)n * DD;
  x[base + t]       = (x[base + t]       - mu) * rs * lw[t]       + lb[t];
  x[base + t + 256] = (x[base + t + 256] - mu) * rs * lw[t + 256] + lb[t + 256];
}

// ---------------- pooling ----------------
__device__ __forceinline__ void atomicMaxF(float* addr, float val) {
  if (val >= 0.0f) atomicMax((int*)addr, __float_as_int(val));
  else             atomicMin((unsigned int*)addr, __float_as_uint(val));
}
__global__ void k_pool(const float* __restrict__ x, const int* __restrict__ batch,
                       float* __restrict__ meanp, float* __restrict__ maxp) {
  int n = blockIdx.x, t = threadIdx.x;
  int g = batch[n];
  size_t base = (size_t)n * DD;
  float v0 = x[base + t], v1 = x[base + t + 256];
  atomicAdd(&meanp[g * DD + t], v0);
  atomicAdd(&meanp[g * DD + t + 256], v1);
  atomicMaxF(&maxp[g * DD + t], v0);
  atomicMaxF(&maxp[g * DD + t + 256], v1);
}
__global__ void k_pool_fin(const float* __restrict__ meanp, const float* __restrict__ maxp,
                           const int* __restrict__ gcount, float* __restrict__ hcat) {
  int i = blockIdx.x * blockDim.x + threadIdx.x;
  if (i >= GG * DD) return;
  int g = i / DD, c = i % DD;
  float cnt = fmaxf((float)gcount[g], 1.0f);
  float mx = maxp[i];
  if (!isfinite(mx)) mx = 0.0f;
  hcat[(size_t)g * (2 * DD) + c]      = meanp[i] / cnt;
  hcat[(size_t)g * (2 * DD) + DD + c] = mx;
}

// ---------------- head ----------------
__global__ void k_bias_gelu(float* __restrict__ h, const float* __restrict__ b) {
  int i = blockIdx.x * blockDim.x + threadIdx.x;
  if (i >= GG * 2 * DD) return;
  float v = h[i] + b[i % (2 * DD)];
  h[i] = 0.5f * v * (1.0f + erff(v * 0.70710678118654752f));
}
__global__ void k_norm_out(const float* __restrict__ h2, const float* __restrict__ b,
                           float* __restrict__ out) {
  int w = blockIdx.x * (blockDim.x >> 5) + (threadIdx.x >> 5);
  if (w >= GG) return;
  int lane = threadIdx.x & 31;
  float v[8];
  float ss = 0.f;
  for (int j = 0; j < 8; ++j) {
    int c = lane + j * 32;
    v[j] = h2[(size_t)w * OUTD + c] + b[c];
    ss += v[j] * v[j];
  }
  for (int m = 16; m; m >>= 1) ss += __shfl_xor(ss, m);
  float inv = 1.0f / fmaxf(sqrtf(ss), 1e-12f);
  for (int j = 0; j < 8; ++j) out[(size_t)w * OUTD + lane + j * 32] = v[j] * inv;
}

// ---------------- host orchestration ----------------
static inline void gemm(const float* A, const float* B, float* C, int M, int K, int Nc,
                        hipStream_t s) {
  if (Nc >= 512) {
    dim3 grid(M / 16, Nc / 512);
    k_gemm<512><<<grid, 256, 0, s>>>(A, B, C, M, K, Nc);
  } else {
    dim3 grid(M / 16, Nc / 256);
    k_gemm<256><<<grid, 128, 0, s>>>(A, B, C, M, K, Nc);
  }
}

extern "C" void kernel_launch(void* const* d_in, const int* in_sizes, int n_in,
                              void* d_out, int out_size, void* d_ws, size_t ws_size,
                              hipStream_t stream) {
  const int*   node_ids = (const int*)d_in[0];
  const int*   ei       = (const int*)d_in[1];   // [2,E] flat
  const int*   batch    = (const int*)d_in[2];
  const float* emb      = (const float*)d_in[3];
  const float *W[4], *As[4], *Ad[4], *Bb[4], *Lw[4], *Lb[4];
  for (int i = 0; i < 4; ++i) {
    int b = 4 + i * 6;
    W[i]  = (const float*)d_in[b + 0];
    As[i] = (const float*)d_in[b + 1];
    Ad[i] = (const float*)d_in[b + 2];
    Bb[i] = (const float*)d_in[b + 3];
    Lw[i] = (const float*)d_in[b + 4];
    Lb[i] = (const float*)d_in[b + 5];
  }
  const float* res_w0 = (const float*)d_in[28];
  const float* p1w = (const float*)d_in[29];
  const float* p1b = (const float*)d_in[30];
  const float* p2w = (const float*)d_in[31];
  const float* p2b = (const float*)d_in[32];

  // workspace carving (256B aligned)
  size_t off = 0;
  char* base = (char*)d_ws;
  auto carve = [&](size_t nbytes) -> void* {
    void* p = base + off;
    off += (nbytes + 255) & ~(size_t)255;
    return p;
  };
  float* x0    = (float*)carve((size_t)NN * DE * 4);
  float* bufA  = (float*)carve((size_t)NN * DD * 4);
  float* bufB  = (float*)carve((size_t)NN * DD * 4);
  float* xp    = (float*)carve((size_t)NN * DD * 4);
  float* agg   = (float*)carve((size_t)NN * DD * 4);
  float* ssrc  = (float*)carve((size_t)NN * HH * 4);
  float* sdst  = (float*)carve((size_t)NN * HH * 4);
  int*   counts = (int*)carve((size_t)NN * 4);
  int*   indptr = (int*)carve((size_t)(NN + 1) * 4);
  int*   cursor = (int*)carve((size_t)NN * 4);
  int*   csrs   = (int*)carve((size_t)ETOT * 4);
  int*   gcount = (int*)carve((size_t)GG * 4);
  float* gstats = (float*)carve((size_t)4 * GG * 4);  // [gsum|gsq|gmu|grs]
  float* gsum = gstats, *gsq = gstats + GG, *gmu = gstats + 2 * GG, *grs = gstats + 3 * GG;
  float* meanp = (float*)carve((size_t)GG * DD * 4);
  float* maxp  = (float*)carve((size_t)GG * DD * 4);
  float* hcat  = (float*)carve((size_t)GG * 2 * DD * 4);
  float* h1    = (float*)carve((size_t)GG * 2 * DD * 4);
  float* h2    = (float*)carve((size_t)GG * OUTD * 4);
  (void)ws_size; (void)in_sizes; (void)n_in; (void)out_size;

  // ---- graph structure (once per call) ----
  k_fill_i32<<<(NN + 255) / 256, 256, 0, stream>>>(counts, 0, NN);
  k_fill_i32<<<1, 256, 0, stream>>>(gcount, 0, GG);
  k_count_deg<<<(ETOT + 255) / 256, 256, 0, stream>>>(ei, counts);
  k_scan<<<1, 1024, 0, stream>>>(counts, indptr);
  k_copy_i32<<<(NN + 255) / 256, 256, 0, stream>>>(cursor, indptr, NN);
  k_fill_csr<<<(ETOT + 255) / 256, 256, 0, stream>>>(ei, cursor, csrs);
  k_gcount<<<(NN + 255) / 256, 256, 0, stream>>>(batch, gcount);

  // ---- embedding + layer-0 residual projection ----
  k_gather_emb<<<(NN * DE + 255) / 256, 256, 0, stream>>>(node_ids, emb, x0);
  gemm(x0, res_w0, bufA, NN, DE, DD, stream);   // residual for layer 0 -> bufA

  // ---- 4 GAT layers ----
  for (int i = 0; i < 4; ++i) {
    const float* xin = (i == 0) ? x0 : ((i & 1) ? bufB : bufA);
    const float* res = (i == 0) ? bufA : xin;
    float* dst = (i & 1) ? bufA : bufB;        // L0->bufB, L1->bufA, L2->bufB, L3->bufA
    int K = (i == 0) ? DE : DD;
    gemm(xin, W[i], xp, NN, K, DD, stream);
    k_scores<<<(NN * HH + 7) / 8, 256, 0, stream>>>(xp, As[i], Ad[i], ssrc, sdst);
    k_aggregate<<<NN, 128, 0, stream>>>(xp, ssrc, sdst, indptr, csrs, agg);
    k_fill_f32<<<1, 256, 0, stream>>>(gsum, 0.0f, 2 * GG);  // zero gsum+gsq
    k_combine_stats<<<NN, 256, 0, stream>>>(agg, Bb[i], res, batch, dst, gsum, gsq);
    k_ln_params<<<1, 128, 0, stream>>>(gsum, gsq, gcount, gmu, grs);
    k_ln_apply<<<NN, 256, 0, stream>>>(dst, batch, gmu, grs, Lw[i], Lb[i]);
  }
  float* xfin = bufA;  // layer 3 output

  // ---- pooling ----
  k_fill_f32<<<(GG * DD + 255) / 256, 256, 0, stream>>>(meanp, 0.0f, GG * DD);
  k_fill_f32<<<(GG * DD + 255) / 256, 256, 0, stream>>>(maxp, -INFINITY, GG * DD);
  k_pool<<<NN, 256, 0, stream>>>(xfin, batch, meanp, maxp);
  k_pool_fin<<<(GG * DD + 255) / 256, 256, 0, stream>>>(meanp, maxp, gcount, hcat);

  // ---- MLP head ----
  gemm(hcat, p1w, h1, GG, 2 * DD, 2 * DD, stream);
  k_bias_gelu<<<(GG * 2 * DD + 255) / 256, 256, 0, stream>>>(h1, p1b);
  gemm(h1, p2w, h2, GG, 2 * DD, OUTD, stream);
  k_norm_out<<<(GG + 7) / 8, 256, 0, stream>>>(h2, p2b, (float*)d_out);
}